// GNNActor_12884901888485
// MI455X (gfx1250) — compile-verified
//
#include <hip/hip_runtime.h>

#define N_NODES 100000
#define N_EDGES 1200000
#define IN_CH   64
#define HID     32
#define OUT_CH  2
#define N_TILES (N_NODES / 16)   // 6250, exact

typedef __attribute__((ext_vector_type(2))) float v2f;
typedef __attribute__((ext_vector_type(8))) float v8f;

// ---------------------------------------------------------------------------
// Degree: deg[i] = 1 (self loop) + #edges with dst == i
// ---------------------------------------------------------------------------
__global__ void k_deg_init(float* __restrict__ deg) {
    int i = blockIdx.x * blockDim.x + threadIdx.x;
    if (i < N_NODES) deg[i] = 1.0f;
}

__global__ void k_deg_accum(const long long* __restrict__ dst, float* __restrict__ deg) {
    int e = blockIdx.x * blockDim.x + threadIdx.x;
    if (e < N_EDGES) {
        int d = (int)dst[e];
        (void)__hip_atomic_fetch_add(&deg[d], 1.0f, __ATOMIC_RELAXED,
                                     __HIP_MEMORY_SCOPE_AGENT);
    }
}

__global__ void k_rsqrt_inplace(float* __restrict__ deg) {
    int i = blockIdx.x * blockDim.x + threadIdx.x;
    if (i < N_NODES) deg[i] = rsqrtf(deg[i]);
}

// ---------------------------------------------------------------------------
// h = x @ W_gcn  (100000x64 @ 64x64), fp32 WMMA 16x16x4.
// One wave per 16x16 output tile. 6250 row tiles * 4 col tiles = 25000 waves.
// A frag (16x4): lane l -> row = l&15, holds K = 2*(l>>4), 2*(l>>4)+1
// B frag (4x16): lane l -> col = l&15, holds K = 2*(l>>4), 2*(l>>4)+1
// C/D (16x16):   lane l -> col = l&15, VGPR j holds row M = j + 8*(l>>4)
// ---------------------------------------------------------------------------
__global__ void k_gemm_xw(const float* __restrict__ x, const float* __restrict__ W,
                          float* __restrict__ h) {
    const int lane = threadIdx.x & 31;
    const int wave = blockIdx.x * (blockDim.x >> 5) + (threadIdx.x >> 5);
    const int rowTile = wave >> 2;        // /4 col tiles
    const int colTile = wave & 3;
    if (rowTile >= N_TILES) return;       // wave-uniform
    const int lo = lane & 15;
    const int hi = lane >> 4;
    const int row = rowTile * 16 + lo;
    const int col = colTile * 16 + lo;

    v8f acc = {};
#pragma unroll
    for (int k0 = 0; k0 < IN_CH; k0 += 4) {
        const int k = k0 + 2 * hi;
        v2f a, b;
        a.x = x[row * IN_CH + k];
        a.y = x[row * IN_CH + k + 1];
        b.x = W[k * IN_CH + col];
        b.y = W[(k + 1) * IN_CH + col];
        acc = __builtin_amdgcn_wmma_f32_16x16x4_f32(false, a, false, b,
                                                    (short)0, acc, false, false);
    }
#pragma unroll
    for (int j = 0; j < 8; ++j) {
        const int m = rowTile * 16 + j + 8 * hi;
        h[m * IN_CH + col] = acc[j];
    }
}

// ---------------------------------------------------------------------------
// agg[i] = h[i] * dinv[i]^2   (self-loop contribution), float4 vectorized
// ---------------------------------------------------------------------------
__global__ void k_selfloop(const float* __restrict__ h, const float* __restrict__ dinv,
                           float* __restrict__ agg) {
    int tid = blockIdx.x * blockDim.x + threadIdx.x;
    if (tid >= N_NODES * (IN_CH / 4)) return;
    int i = tid >> 4;            // node
    int c = (tid & 15) * 4;      // channel base
    float s = dinv[i];
    s = s * s;
    float4 v = *(const float4*)(h + (size_t)i * IN_CH + c);
    float4 r = make_float4(v.x * s, v.y * s, v.z * s, v.w * s);
    *(float4*)(agg + (size_t)i * IN_CH + c) = r;
}

// ---------------------------------------------------------------------------
// Edge scatter: agg[dst] += h[src] * dinv[src]*dinv[dst]
// 16 threads per edge, float4 gather, fp32 global atomics (L2-resident).
// ---------------------------------------------------------------------------
__global__ void k_scatter(const float* __restrict__ h, const float* __restrict__ dinv,
                          const long long* __restrict__ src,
                          const long long* __restrict__ dst,
                          float* __restrict__ agg) {
    long long tid = (long long)blockIdx.x * blockDim.x + threadIdx.x;
    if (tid >= (long long)N_EDGES * 16) return;
    int e = (int)(tid >> 4);
    int c = (int)(tid & 15) * 4;
    int s = (int)src[e];
    int d = (int)dst[e];
    float norm = dinv[s] * dinv[d];
    float4 v = *(const float4*)(h + (size_t)s * IN_CH + c);
    float* ap = agg + (size_t)d * IN_CH + c;
    (void)__hip_atomic_fetch_add(ap + 0, v.x * norm, __ATOMIC_RELAXED, __HIP_MEMORY_SCOPE_AGENT);
    (void)__hip_atomic_fetch_add(ap + 1, v.y * norm, __ATOMIC_RELAXED, __HIP_MEMORY_SCOPE_AGENT);
    (void)__hip_atomic_fetch_add(ap + 2, v.z * norm, __ATOMIC_RELAXED, __HIP_MEMORY_SCOPE_AGENT);
    (void)__hip_atomic_fetch_add(ap + 3, v.w * norm, __ATOMIC_RELAXED, __HIP_MEMORY_SCOPE_AGENT);
}

// ---------------------------------------------------------------------------
// Fused residual + MLP. One wave per 16-node tile.
//   hres = relu(agg + b_gcn) + x             (computed into A frags on the fly)
//   h1   = relu(hres @ W1 + b1)   [16x32]    WMMA, staged in LDS
//   h2   = relu(h1   @ W2 + b2)   [16x32]    WMMA, staged in LDS
//   y    =       h2  @ W3 + b3    [16x2]     WMMA with zero-padded B
// ---------------------------------------------------------------------------
#define LPAD 33   // LDS row stride (floats), padded to dodge bank conflicts

__global__ void k_fused_mlp(const float* __restrict__ agg, const float* __restrict__ x,
                            const float* __restrict__ bg,
                            const float* __restrict__ W1, const float* __restrict__ b1,
                            const float* __restrict__ W2, const float* __restrict__ b2,
                            const float* __restrict__ W3, const float* __restrict__ b3,
                            float* __restrict__ out) {
    __shared__ float h1s[8][16 * LPAD];
    __shared__ float h2s[8][16 * LPAD];
    const int lane = threadIdx.x & 31;
    const int w    = threadIdx.x >> 5;
    const int t    = blockIdx.x * 8 + w;
    const bool active = (t < N_TILES);      // wave-uniform
    const int tile = active ? t : 0;
    const int lo = lane & 15;
    const int hi = lane >> 4;
    const int rowBase = tile * 16;
    const int row = rowBase + lo;

    // ---- stage 1: h1 = relu( (relu(agg+bg)+x) @ W1 + b1 ), K=64, 2 col tiles
    v8f acc1[2] = {v8f{}, v8f{}};
#pragma unroll
    for (int k0 = 0; k0 < IN_CH; k0 += 4) {
        const int k = k0 + 2 * hi;
        float a0 = fmaxf(agg[row * IN_CH + k]     + bg[k],     0.0f) + x[row * IN_CH + k];
        float a1 = fmaxf(agg[row * IN_CH + k + 1] + bg[k + 1], 0.0f) + x[row * IN_CH + k + 1];
        v2f a; a.x = a0; a.y = a1;
#pragma unroll
        for (int ct = 0; ct < 2; ++ct) {
            v2f b;
            b.x = W1[k * HID + ct * 16 + lo];
            b.y = W1[(k + 1) * HID + ct * 16 + lo];
            acc1[ct] = __builtin_amdgcn_wmma_f32_16x16x4_f32(false, a, false, b,
                                                             (short)0, acc1[ct], false, false);
        }
    }
#pragma unroll
    for (int ct = 0; ct < 2; ++ct) {
        float bb = b1[ct * 16 + lo];
#pragma unroll
        for (int j = 0; j < 8; ++j) {
            int m = j + 8 * hi;
            h1s[w][m * LPAD + ct * 16 + lo] = fmaxf(acc1[ct][j] + bb, 0.0f);
        }
    }
    __syncthreads();

    // ---- stage 2: h2 = relu(h1 @ W2 + b2), K=32, 2 col tiles
    v8f acc2[2] = {v8f{}, v8f{}};
#pragma unroll
    for (int k0 = 0; k0 < HID; k0 += 4) {
        const int k = k0 + 2 * hi;
        v2f a;
        a.x = h1s[w][lo * LPAD + k];
        a.y = h1s[w][lo * LPAD + k + 1];
#pragma unroll
        for (int ct = 0; ct < 2; ++ct) {
            v2f b;
            b.x = W2[k * HID + ct * 16 + lo];
            b.y = W2[(k + 1) * HID + ct * 16 + lo];
            acc2[ct] = __builtin_amdgcn_wmma_f32_16x16x4_f32(false, a, false, b,
                                                             (short)0, acc2[ct], false, false);
        }
    }
#pragma unroll
    for (int ct = 0; ct < 2; ++ct) {
        float bb = b2[ct * 16 + lo];
#pragma unroll
        for (int j = 0; j < 8; ++j) {
            int m = j + 8 * hi;
            h2s[w][m * LPAD + ct * 16 + lo] = fmaxf(acc2[ct][j] + bb, 0.0f);
        }
    }
    __syncthreads();

    // ---- stage 3: y = h2 @ W3 + b3, K=32, W3 padded to 16 cols (only 2 valid)
    v8f acc3 = {};
#pragma unroll
    for (int k0 = 0; k0 < HID; k0 += 4) {
        const int k = k0 + 2 * hi;
        v2f a;
        a.x = h2s[w][lo * LPAD + k];
        a.y = h2s[w][lo * LPAD + k + 1];
        v2f b;
        b.x = (lo < OUT_CH) ? W3[k * OUT_CH + lo]       : 0.0f;
        b.y = (lo < OUT_CH) ? W3[(k + 1) * OUT_CH + lo] : 0.0f;
        acc3 = __builtin_amdgcn_wmma_f32_16x16x4_f32(false, a, false, b,
                                                     (short)0, acc3, false, false);
    }
    if (active && lo < OUT_CH) {
        float bb = b3[lo];
#pragma unroll
        for (int j = 0; j < 8; ++j) {
            int m = rowBase + j + 8 * hi;
            out[m * OUT_CH + lo] = acc3[j] + bb;
        }
    }
}

// ---------------------------------------------------------------------------
extern "C" void kernel_launch(void* const* d_in, const int* in_sizes, int n_in,
                              void* d_out, int out_size, void* d_ws, size_t ws_size,
                              hipStream_t stream) {
    const float*     x   = (const float*)d_in[0];
    const long long* ei  = (const long long*)d_in[1];   // int64 edge_index [2, E]
    const float*     Wg  = (const float*)d_in[2];
    const float*     bg  = (const float*)d_in[3];
    const float*     W1  = (const float*)d_in[4];
    const float*     b1  = (const float*)d_in[5];
    const float*     W2  = (const float*)d_in[6];
    const float*     b2  = (const float*)d_in[7];
    const float*     W3  = (const float*)d_in[8];
    const float*     b3  = (const float*)d_in[9];
    float*           out = (float*)d_out;

    const long long* src = ei;
    const long long* dst = ei + N_EDGES;

    // workspace layout: deg/dinv [N] | h [N*64] | agg [N*64]  (~51.6 MB)
    float* ws  = (float*)d_ws;
    float* deg = ws;
    float* h   = ws + N_NODES;
    float* agg = h + (size_t)N_NODES * IN_CH;

    k_deg_init<<<(N_NODES + 255) / 256, 256, 0, stream>>>(deg);
    k_deg_accum<<<(N_EDGES + 255) / 256, 256, 0, stream>>>(dst, deg);
    k_rsqrt_inplace<<<(N_NODES + 255) / 256, 256, 0, stream>>>(deg);

    // 25000 waves, 8 waves per 256-thread block
    k_gemm_xw<<<(N_TILES * 4 + 7) / 8, 256, 0, stream>>>(x, Wg, h);

    k_selfloop<<<(N_NODES * (IN_CH / 4) + 255) / 256, 256, 0, stream>>>(h, deg, agg);

    long long scat_threads = (long long)N_EDGES * 16;
    k_scatter<<<(int)((scat_threads + 255) / 256), 256, 0, stream>>>(h, deg, src, dst, agg);

    k_fused_mlp<<<(N_TILES + 7) / 8, 256, 0, stream>>>(agg, x, bg, W1, b1, W2, b2, W3, b3, out);
}